// QNN_0_42889543418121
// MI455X (gfx1250) — compile-verified
//
#include <hip/hip_runtime.h>
#include <hip/hip_bf16.h>
#include <math.h>

// ---------------------------------------------------------------------------
// QNN forward: per-element exp(i*H) via closed-form 3x3 Hermitian eigensolver
// (traceless, trigonometric eigenvalues + Cayley-Hamilton reconstruction),
// then 8 layers of [U_enc, U_var[l]] complex 3x3 matvecs, then normalization.
// VALU-bound (~2.7 GFLOP f32); memory floor ~1.9us (44 MB @ 23.3 TB/s).
// CDNA5 paths: async global->LDS staging of U_var, global_prefetch_b8,
// packed-f32 (v_pk_*) dual-rate VALU. WMMA inapplicable: 3x3 complex chains
// with per-lane-distinct matrices and sequential layer dependency.
// NOTE: IEEE divides and sqrtf are kept rather than raw v_rcp/v_sqrt builtins.
// ---------------------------------------------------------------------------

#if defined(__has_builtin)
#if __has_builtin(__builtin_amdgcn_global_load_async_to_lds_b32) && \
    __has_builtin(__builtin_amdgcn_s_wait_asynccnt)
#define QNN_ASYNC_LDS 1
#endif
#endif
#ifndef QNN_ASYNC_LDS
#define QNN_ASYNC_LDS 0
#endif

#define QNN_AS1 __attribute__((address_space(1)))
#define QNN_AS3 __attribute__((address_space(3)))

// exp(i*H) for traceless Hermitian 3x3 H = sum_k c[k] * gellmann[k].
// U laid out row-major complex: U[(i*3+j)*2 + {0:re,1:im}].
__device__ __forceinline__ void expm_iH(const float* c, float* U) {
  const float s3 = 0.57735026918962576f; // 1/sqrt(3)
  float h00 = c[2] + c[7] * s3;
  float h11 = c[7] * s3 - c[2];
  float h22 = -2.0f * c[7] * s3;
  float ar = c[0], ai = -c[1];   // h01
  float br = c[3], bi = -c[4];   // h02
  float cr = c[5], ci = -c[6];   // h12

  float a2 = ar * ar + ai * ai;
  float b2 = br * br + bi * bi;
  float c2 = cr * cr + ci * ci;

  // Trigonometric eigenvalues (trace(H)=0 -> q=0)
  float p2 = h00 * h00 + h11 * h11 + h22 * h22 + 2.0f * (a2 + b2 + c2);
  float p  = sqrtf(p2 * (1.0f / 6.0f) + 1e-30f);
  float invp = 1.0f / p;

  float acr = ar * cr - ai * ci;          // (a*c).re
  float aci = ar * ci + ai * cr;          // (a*c).im
  float det = h00 * h11 * h22 - h00 * c2 - h22 * a2 - h11 * b2
            + 2.0f * (acr * br + aci * bi);   // + 2*Re(a*c*conj(b))

  float r = 0.5f * det * invp * invp * invp;
  r = fminf(1.0f, fmaxf(-1.0f, r));
  float phi = acosf(r) * (1.0f / 3.0f);
  float twop = 2.0f * p;

  // l0 = 2p*cos(phi); l1 = 2p*cos(phi + 2pi/3) = 2p*(-0.5*cos - (sqrt3/2)*sin)
  float sphi, cphi;
  sincosf(phi, &sphi, &cphi);
  float l0 = twop * cphi;
  float l1 = twop * (-0.5f * cphi - 0.86602540378443865f * sphi);
  float l2 = -l0 - l1;

  // H^2 (Hermitian)
  float m00 = h00 * h00 + a2 + b2;
  float m11 = a2 + h11 * h11 + c2;
  float m22 = b2 + c2 + h22 * h22;
  float t01 = h00 + h11, t02 = h00 + h22, t12 = h11 + h22;
  float m01r = ar * t01 + (br * cr + bi * ci);     // a*(h00+h11) + b*conj(c)
  float m01i = ai * t01 + (bi * cr - br * ci);
  float m02r = br * t02 + acr;                     // b*(h00+h22) + a*c
  float m02i = bi * t02 + aci;
  float m12r = cr * t12 + (ar * br + ai * bi);     // c*(h11+h22) + conj(a)*b
  float m12i = ci * t12 + (ar * bi - ai * br);

  // e^{i*l0}, e^{i*l1} via sincos; e^{i*l2} = e^{-i(l0+l1)} via identity
  float s0, c0, s1, c1;
  sincosf(l0, &s0, &c0);
  sincosf(l1, &s1, &c1);
  float c2v = c0 * c1 - s0 * s1;          // cos(l0+l1) = cos(-l2)
  float s2v = -(s0 * c1 + c0 * s1);       // sin(l2) = -sin(l0+l1)

  // Cayley-Hamilton: exp(iH) = A*H^2 + B*H + C*I
  // A = sum e_k, B = sum e_k*l_k, C = sum e_k*(l_j*l_l), e_k = e^{i l_k}/d_k
  float Ar = 0.f, Ai = 0.f, Br = 0.f, Bi = 0.f, Cr = 0.f, Ci = 0.f;
  float lam[3] = {l0, l1, l2};
  float cc[3]  = {c0, c1, c2v};
  float ss[3]  = {s0, s1, s2v};
#pragma unroll
  for (int k = 0; k < 3; ++k) {
    float lk = lam[k];
    float lj = lam[k == 0 ? 1 : (k == 1 ? 2 : 0)];
    float ll = lam[k == 0 ? 2 : (k == 1 ? 0 : 1)];
    float dk = (lk - lj) * (lk - ll);
    float ad = fmaxf(fabsf(dk), 1e-18f);            // degenerate-eig guard
    dk = (dk < 0.f) ? -ad : ad;
    float inv = 1.0f / dk;
    float er = cc[k] * inv, ei = ss[k] * inv;
    float mu = lj * ll;
    Ar += er;       Ai += ei;
    Br += er * lk;  Bi += ei * lk;
    Cr += er * mu;  Ci += ei * mu;
  }

  // Assemble U (m, h Hermitian; A,B,C complex)
  U[0]  = Ar * m00 + Br * h00 + Cr;  U[1]  = Ai * m00 + Bi * h00 + Ci;  // U00
  U[8]  = Ar * m11 + Br * h11 + Cr;  U[9]  = Ai * m11 + Bi * h11 + Ci;  // U11
  U[16] = Ar * m22 + Br * h22 + Cr;  U[17] = Ai * m22 + Bi * h22 + Ci;  // U22
  U[2]  = Ar * m01r - Ai * m01i + Br * ar - Bi * ai;                    // U01
  U[3]  = Ar * m01i + Ai * m01r + Br * ai + Bi * ar;
  U[6]  = Ar * m01r + Ai * m01i + Br * ar + Bi * ai;                    // U10
  U[7]  = Ai * m01r - Ar * m01i + Bi * ar - Br * ai;
  U[4]  = Ar * m02r - Ai * m02i + Br * br - Bi * bi;                    // U02
  U[5]  = Ar * m02i + Ai * m02r + Br * bi + Bi * br;
  U[12] = Ar * m02r + Ai * m02i + Br * br + Bi * bi;                    // U20
  U[13] = Ai * m02r - Ar * m02i + Bi * br - Br * bi;
  U[10] = Ar * m12r - Ai * m12i + Br * cr - Bi * ci;                    // U12
  U[11] = Ar * m12i + Ai * m12r + Br * ci + Bi * cr;
  U[14] = Ar * m12r + Ai * m12i + Br * cr + Bi * ci;                    // U21
  U[15] = Ai * m12r - Ar * m12i + Bi * cr - Br * ci;
}

// complex 3x3 matvec: x <- M x, M row-major complex (18 floats)
__device__ __forceinline__ void mv3(const float* M,
                                    float& x0r, float& x0i,
                                    float& x1r, float& x1i,
                                    float& x2r, float& x2i) {
  float y0r = M[0]*x0r - M[1]*x0i + M[2]*x1r - M[3]*x1i + M[4]*x2r - M[5]*x2i;
  float y0i = M[0]*x0i + M[1]*x0r + M[2]*x1i + M[3]*x1r + M[4]*x2i + M[5]*x2r;
  float y1r = M[6]*x0r - M[7]*x0i + M[8]*x1r - M[9]*x1i + M[10]*x2r - M[11]*x2i;
  float y1i = M[6]*x0i + M[7]*x0r + M[8]*x1i + M[9]*x1r + M[10]*x2i + M[11]*x2r;
  float y2r = M[12]*x0r - M[13]*x0i + M[14]*x1r - M[15]*x1i + M[16]*x2r - M[17]*x2i;
  float y2i = M[12]*x0i + M[13]*x0r + M[14]*x1i + M[15]*x1r + M[16]*x2i + M[17]*x2r;
  x0r = y0r; x0i = y0i; x1r = y1r; x1i = y1i; x2r = y2r; x2i = y2i;
}

// Pre-kernel: U_var[l] = exp(i * sum_p weights[l,p]*G_p), 8 matrices -> d_ws
__global__ void qnn_prep(const float* __restrict__ weights,
                         float* __restrict__ uvar) {
  int l = threadIdx.x;
  if (l < 8) {
    float c[8];
#pragma unroll
    for (int p = 0; p < 8; ++p) c[p] = weights[l * 8 + p];
    float U[18];
    expm_iH(c, U);
#pragma unroll
    for (int k = 0; k < 18; ++k) uvar[l * 18 + k] = U[k];
  }
}

__global__ __launch_bounds__(256) void qnn_main(const float* __restrict__ batch,
                                                const float* __restrict__ uvar,
                                                float* __restrict__ out,
                                                int nB) {
  __shared__ float sUv[144]; // 8 layers x 3x3 complex
  const int t = threadIdx.x;

  // Stage shared U_var table into LDS (CDNA5 async global->LDS)
#if QNN_ASYNC_LDS
  if (t < 144) {
    float* uvnc = const_cast<float*>(uvar);
    __builtin_amdgcn_global_load_async_to_lds_b32(
        (QNN_AS1 int*)(uvnc + t), (QNN_AS3 int*)(sUv + t), 0, 0);
  }
  __builtin_amdgcn_s_wait_asynccnt(0);
#else
  if (t < 144) sUv[t] = uvar[t];
#endif
  __syncthreads();

  const int b = blockIdx.x * 256 + t;
  if (b >= nB) return;

#if defined(__gfx1250__)
  __builtin_prefetch(batch + (size_t)b * 8, 0, 3); // global_prefetch_b8
#endif

  // 32B contiguous per lane -> two global_load_b128
  const float4* bp = (const float4*)(batch);
  float4 q0 = bp[(size_t)b * 2 + 0];
  float4 q1 = bp[(size_t)b * 2 + 1];
  float cv[8] = {q0.x, q0.y, q0.z, q0.w, q1.x, q1.y, q1.z, q1.w};

  float Ue[18];
  expm_iH(cv, Ue);

  // psi0 = e0 -> U_enc @ psi0 = column 0 of U_enc
  float x0r = Ue[0],  x0i = Ue[1];
  float x1r = Ue[6],  x1i = Ue[7];
  float x2r = Ue[12], x2i = Ue[13];
  mv3(sUv, x0r, x0i, x1r, x1i, x2r, x2i);            // layer 0 variational

#pragma unroll
  for (int l = 1; l < 8; ++l) {
    mv3(Ue, x0r, x0i, x1r, x1i, x2r, x2i);           // encoding
    mv3(sUv + l * 18, x0r, x0i, x1r, x1i, x2r, x2i); // variational (LDS bcast)
  }

  float d0 = x0r * x0r + x0i * x0i;
  float d1 = x1r * x1r + x1i * x1i;
  float d2 = x2r * x2r + x2i * x2i;
  float f0 = sqrtf(fmaxf(d0, 1e-10f)) + 2e-5f; f0 *= f0;
  float f1 = sqrtf(fmaxf(d1, 1e-10f)) + 2e-5f; f1 *= f1;
  float f2 = sqrtf(fmaxf(d2, 1e-10f)) + 2e-5f; f2 *= f2;
  float s = 1.0f / (f0 + f1 + f2);

  size_t o = (size_t)b * 3;
  out[o + 0] = f0 * s;
  out[o + 1] = f1 * s;
  out[o + 2] = f2 * s;
}

extern "C" void kernel_launch(void* const* d_in, const int* in_sizes, int n_in,
                              void* d_out, int out_size, void* d_ws, size_t ws_size,
                              hipStream_t stream) {
  const float* batch   = (const float*)d_in[0];   // [B, 8] f32
  const float* weights = (const float*)d_in[1];   // [8, 8] f32
  float* out  = (float*)d_out;                    // [B, 3] f32
  float* uvar = (float*)d_ws;                     // 8 * 18 floats scratch

  const int nB = in_sizes[0] / 8;

  qnn_prep<<<1, 32, 0, stream>>>(weights, uvar);
  qnn_main<<<(nB + 255) / 256, 256, 0, stream>>>(batch, uvar, out, nB);
}